// TopKActivation_13151189861106
// MI455X (gfx1250) — compile-verified
//
#include <hip/hip_runtime.h>

#define NCOLS     32768
#define NTHREADS  1024
#define TOPK      64
#define NBINS     4096
#define CAND_CAP  1024

typedef int v4i __attribute__((ext_vector_type(4)));
typedef __attribute__((address_space(1))) v4i* global_b128_ptr;
typedef __attribute__((address_space(3))) v4i* lds_b128_ptr;

// Monotonic float -> uint key: preserves total order of IEEE floats.
__device__ __forceinline__ unsigned orderKey(float f) {
    unsigned u = __float_as_uint(f);
    return u ^ ((unsigned)((int)u >> 31) | 0x80000000u);
}

// Block-wide inclusive Hillis-Steele scan over NTHREADS entries.
__device__ __forceinline__ void inclusiveScan(unsigned* buf, int tid) {
    #pragma unroll 1
    for (int off = 1; off < NTHREADS; off <<= 1) {
        unsigned add = (tid >= off) ? buf[tid - off] : 0u;
        __syncthreads();
        buf[tid] += add;
        __syncthreads();
    }
}

extern "C" __global__ __launch_bounds__(NTHREADS, 1)
void topk_relu_rowselect(const float* __restrict__ x, float* __restrict__ out) {
    // LDS: 128KB row + 16KB hist + 4KB scan + 4KB candidates ~= 152KB (WGP has 320KB)
    __shared__ __align__(16) float rowBuf[NCOLS];
    __shared__ unsigned hist[NBINS];
    __shared__ unsigned scanBuf[NTHREADS];
    __shared__ unsigned candKey[CAND_CAP];
    __shared__ unsigned sCandCount;
    __shared__ unsigned sBin, sCumBefore;
    __shared__ unsigned sTKey, sGtAll, sTieTotal;

    const int tid = threadIdx.x;
    const long row = blockIdx.x;
    const float* src = x + row * (long)NCOLS;
    float*       dst = out + row * (long)NCOLS;

    // ---- Stage the row Global -> LDS via gfx1250 async copy (ASYNCcnt path) ----
#if __has_builtin(__builtin_amdgcn_global_load_async_to_lds_b128)
    {
        #pragma unroll
        for (int it = 0; it < NCOLS / (NTHREADS * 4); ++it) {
            int e = (tid + it * NTHREADS) * 4;   // 4 floats = 16B per lane per issue
            __builtin_amdgcn_global_load_async_to_lds_b128(
                (global_b128_ptr)(src + e),
                (lds_b128_ptr)(&rowBuf[e]),
                /*imm offset*/0, /*cpol*/0);
        }
    }
#else
    {
        #pragma unroll
        for (int it = 0; it < NCOLS / (NTHREADS * 4); ++it) {
            int e = (tid + it * NTHREADS) * 4;
            *(float4*)(&rowBuf[e]) = *(const float4*)(src + e);
        }
    }
#endif

    // Zero histogram + counters while the async copies are in flight.
    #pragma unroll
    for (int i = tid; i < NBINS; i += NTHREADS) hist[i] = 0u;
    if (tid == 0) { sCandCount = 0u; }

#if __has_builtin(__builtin_amdgcn_s_wait_asynccnt)
    __builtin_amdgcn_s_wait_asynccnt(0);
#elif __has_builtin(__builtin_amdgcn_global_load_async_to_lds_b128)
    asm volatile("s_wait_asynccnt 0" ::: "memory");
#endif
    __syncthreads();

    // ---- Pass 1: 12-bit key histogram (LDS atomics, conflict-free strided reads) ----
    #pragma unroll 4
    for (int i = tid; i < NCOLS; i += NTHREADS) {
        atomicAdd(&hist[orderKey(rowBuf[i]) >> 20], 1u);
    }
    __syncthreads();

    // ---- Find the bin where cumulative-from-top count crosses TOPK ----
    // Descending index d = 0..4095 maps to bin 4095-d. Thread g owns d in [4g, 4g+3].
    unsigned c0 = hist[4095 - 4 * tid];
    unsigned c1 = hist[4094 - 4 * tid];
    unsigned c2 = hist[4093 - 4 * tid];
    unsigned c3 = hist[4092 - 4 * tid];
    unsigned gsum = c0 + c1 + c2 + c3;
    scanBuf[tid] = gsum;
    __syncthreads();
    inclusiveScan(scanBuf, tid);
    {
        unsigned run = scanBuf[tid] - gsum;      // exclusive prefix: counts in bins above
        unsigned cs[4] = {c0, c1, c2, c3};
        #pragma unroll
        for (int j = 0; j < 4; ++j) {
            unsigned c = cs[j];
            if (run < TOPK && run + c >= TOPK) { // unique crossing point
                sBin = 4095u - (4u * tid + j);
                sCumBefore = run;                // # elements strictly above this bin
            }
            run += c;
        }
    }
    __syncthreads();

    // ---- Pass 2: gather candidate keys from the threshold bin ----
    const unsigned tbin = sBin;
    #pragma unroll 4
    for (int i = tid; i < NCOLS; i += NTHREADS) {
        unsigned key = orderKey(rowBuf[i]);
        if ((key >> 20) == tbin) {
            unsigned p = atomicAdd(&sCandCount, 1u);
            if (p < CAND_CAP) candKey[p] = key;
        }
    }
    __syncthreads();

    // ---- Exact select: budget-th largest candidate key (O(n^2) counting, n ~ 8) ----
    const unsigned budget = TOPK - sCumBefore;   // in [1, 64]
    const unsigned n = min(sCandCount, (unsigned)CAND_CAP);
    if (tid < (int)n) {
        unsigned ki = candKey[tid];
        unsigned cgt = 0, cge = 0;
        for (unsigned j = 0; j < n; ++j) {
            unsigned kj = candKey[j];
            cgt += (kj > ki);
            cge += (kj >= ki);
        }
        if (cgt < budget && budget <= cge) {     // ki is the budget-th largest
            sTKey = ki;                          // (ties write identical values)
            sGtAll = sCumBefore + cgt;           // total strictly-greater in row
            sTieTotal = cge - cgt;               // total equal-to-threshold in row
        }
    }
    __syncthreads();

    const unsigned tkey = sTKey;
    const unsigned tieBudget = TOPK - sGtAll;

    if (sTieTotal <= tieBudget) {
        // ---- Fast path (no excess ties): fully coalesced float4 stream-out ----
        const float4* r4 = (const float4*)rowBuf;
        float4*       o4 = (float4*)dst;
        #pragma unroll
        for (int it = 0; it < NCOLS / (NTHREADS * 4); ++it) {
            int i4 = tid + it * NTHREADS;
            float4 v = r4[i4];
            float4 o;
            o.x = (orderKey(v.x) >= tkey) ? fmaxf(v.x, 0.f) : 0.f;
            o.y = (orderKey(v.y) >= tkey) ? fmaxf(v.y, 0.f) : 0.f;
            o.z = (orderKey(v.z) >= tkey) ? fmaxf(v.z, 0.f) : 0.f;
            o.w = (orderKey(v.w) >= tkey) ? fmaxf(v.w, 0.f) : 0.f;
            o4[i4] = o;
        }
    } else {
        // ---- Rare path: deterministic first-index tie-break (matches top_k) ----
        const int base = tid * (NCOLS / NTHREADS);   // contiguous 32-elem segment
        unsigned cnt = 0;
        #pragma unroll
        for (int j = 0; j < NCOLS / NTHREADS; ++j)
            cnt += (orderKey(rowBuf[base + j]) == tkey);
        scanBuf[tid] = cnt;
        __syncthreads();
        inclusiveScan(scanBuf, tid);
        unsigned prefix = scanBuf[tid] - cnt;        // ties at smaller indices
        #pragma unroll
        for (int j = 0; j < NCOLS / NTHREADS; ++j) {
            int i = base + j;
            float f = rowBuf[i];
            unsigned key = orderKey(f);
            float o = 0.f;
            if (key > tkey) {
                o = fmaxf(f, 0.f);
            } else if (key == tkey) {
                if (prefix < tieBudget) o = fmaxf(f, 0.f);
                prefix++;
            }
            dst[i] = o;
        }
    }
}

extern "C" void kernel_launch(void* const* d_in, const int* in_sizes, int n_in,
                              void* d_out, int out_size, void* d_ws, size_t ws_size,
                              hipStream_t stream) {
    const float* x = (const float*)d_in[0];
    float* out = (float*)d_out;
    int rows = in_sizes[0] / NCOLS;   // 8192
    if (rows <= 0) return;
    topk_relu_rowselect<<<rows, NTHREADS, 0, stream>>>(x, out);
}